// ImagePatcherOverlapRandom_65609920414052
// MI455X (gfx1250) — compile-verified
//
#include <hip/hip_runtime.h>
#include <stdint.h>

// ImagePatcherOverlapRandom for MI455X (gfx1250):
//   - async global->LDS DMA (B128, 16B-aligned both sides) of the 3x33x36 window
//   - separable bilinear: VALU column pass, then WMMA f32 16x16x4 row-weight GEMM
//   - LDS transpose + float4 coalesced output stores
// B=64, C=3, S=384, P=32, R=169, scale = 383/384.

typedef __attribute__((ext_vector_type(2))) float v2f;
typedef __attribute__((ext_vector_type(8))) float v8f;

#define NB   64
#define NC   3
#define NS   384
#define NP   32
#define NR   169
#define FSCALE (383.0f / 384.0f)

__device__ __forceinline__ void async_ld_b128(uint32_t lds_off, uint64_t gaddr) {
  // CDNA5 async copy: copies 16 bytes from the global address gaddr to the LDS address lds_off; tracked by ASYNCcnt.
  asm volatile("global_load_async_to_lds_b128 %0, %1, off"
               :: "v"(lds_off), "v"(gaddr)
               : "memory");
}

__global__ __launch_bounds__(128) void
ImagePatcherOverlapRandom_65609920414052_kernel(const float* __restrict__ x,
                                                const int* __restrict__ crop_top,
                                                const int* __restrict__ crop_left,
                                                float* __restrict__ out) {
  __shared__ float s_win[NC][33][36];   // source window: 36 cols from cx4 (16B aligned)
  __shared__ float s_brow[32][36];      // row-weight matrix (q x K), K padded to 36
  __shared__ float s_cint[36][32];      // column-interpolated tile (K x p)
  __shared__ float s_out[NC][32][36];   // transposed result staging, 16B-aligned rows
  __shared__ float s_wx[32];
  __shared__ int   s_x0l[32];

  const int tid  = threadIdx.x;
  const int lane = tid & 31;
  const int wave = tid >> 5;
  const int br   = blockIdx.x;          // 0 .. B*R-1
  const int b    = br / NR;

  const int t = crop_top[br];           // indexes image columns (last axis)
  const int l = crop_left[br];          // indexes image rows
  const int cx  = (int)floorf((float)t * FSCALE);
  const int cx4 = cx & ~3;              // 16B-aligned window start; cx4+35 <= 383
  const int cy  = (int)floorf((float)l * FSCALE);

  // ---- Phase A: async B128 DMA of the 3 x 33 x 36 source window into LDS ----
  {
    const float* imgb = x + (size_t)b * (NC * NS * NS);
    // 99 rows x 9 segments of 4 floats = 891 lane-units, 16B aligned everywhere.
    for (int u = tid; u < NC * 33 * 9; u += 128) {
      int row = u / 9, seg = u - row * 9;
      int c = row / 33, j = row - c * 33;
      const float* g = imgb + ((size_t)c * NS + (size_t)(cy + j)) * NS + cx4 + seg * 4;
      async_ld_b128((uint32_t)(uintptr_t)&s_win[c][j][seg * 4],
                    (uint64_t)(uintptr_t)g);
    }
  }

  // ---- Overlap: build weights while the DMA is in flight -------------------
  if (tid < 32) {                       // column weights (p axis, from crop_top)
    float xp = (float)(tid + t) * FSCALE;
    float x0 = floorf(xp);
    s_wx[tid]  = xp - x0;
    s_x0l[tid] = (int)x0 - cx4;         // in [0,34]; +1 stays inside 36-wide window
  } else if (tid < 64) {                // row-weight matrix (q axis, from crop_left)
    int q = tid - 32;
    float yp = (float)(q + l) * FSCALE;
    float y0 = floorf(yp);
    float wy = yp - y0;
    int y0l = (int)y0 - cy;
    #pragma unroll
    for (int j = 0; j < 36; ++j) s_brow[q][j] = 0.0f;
    s_brow[q][y0l]     = 1.0f - wy;
    s_brow[q][y0l + 1] = wy;
  } else {                              // zero the K-pad rows 33..35 of cint
    int e = tid - 64;                   // 0..63
    s_cint[33 + (e >> 5)][e & 31] = 0.0f;
    if (e < 32) s_cint[35][e] = 0.0f;
  }

  asm volatile("s_wait_asynccnt 0" ::: "memory");
  __syncthreads();

  // ---- Per channel: VALU column pass, then WMMA row-weight GEMM ------------
  const int tm = wave >> 1;             // output tile row (q)
  const int tn = wave & 1;              // output tile col (p)
  const int m  = lane & 15;
  const int kh = (lane >> 4) << 1;      // 0 or 2: K sub-slot by lane half

  for (int c = 0; c < NC; ++c) {
    for (int e = tid; e < 33 * 32; e += 128) {
      int j = e >> 5, p = e & 31;
      float w  = s_wx[p];
      int   i0 = s_x0l[p];
      s_cint[j][p] = (1.0f - w) * s_win[c][j][i0] + w * s_win[c][j][i0 + 1];
    }
    __syncthreads();

    // out'(32x32) = Brow(32x36) x cint(36x32), one 16x16 tile per wave.
    v8f acc = {0.f, 0.f, 0.f, 0.f, 0.f, 0.f, 0.f, 0.f};
    #pragma unroll
    for (int k4 = 0; k4 < 36; k4 += 4) {
      v2f a, bb;
      // A 16x4 f32 fragment: lane M = lane&15; VGPR0 = K{0,2}, VGPR1 = K{1,3}
      a.x  = s_brow[tm * 16 + m][k4 + kh];
      a.y  = s_brow[tm * 16 + m][k4 + kh + 1];
      // B 4x16 f32 fragment: lane N = lane&15; same K slotting
      bb.x = s_cint[k4 + kh][tn * 16 + m];
      bb.y = s_cint[k4 + kh + 1][tn * 16 + m];
      acc = __builtin_amdgcn_wmma_f32_16x16x4_f32(
          /*neg_a=*/false, a, /*neg_b=*/false, bb,
          /*c_mod=*/(short)0, acc, /*reuse_a=*/false, /*reuse_b=*/false);
    }

    // D layout: VGPR i holds M = i + 8*(lane>>4), N = lane&15.  M==q, N==p.
    #pragma unroll
    for (int i = 0; i < 8; ++i) {
      int q = tm * 16 + i + ((lane >> 4) << 3);
      int p = tn * 16 + m;
      s_out[c][p][q] = acc[i];
    }
    __syncthreads();                    // protect s_cint for next channel
  }

  // ---- Coalesced float4 store of the whole 3x32x32 patch -------------------
  float* outp = out + (size_t)br * (NC * NP * NP);
  #pragma unroll
  for (int u = tid; u < (NC * NP * NP) / 4; u += 128) {   // 768 units, 6 full iters
    int c  = u >> 8;
    int p  = (u >> 3) & 31;
    int qw = u & 7;
    float4 v = *(const float4*)&s_out[c][p][qw << 2];
    *(float4*)(outp + ((size_t)u << 2)) = v;
  }
  if (tid == 0) {
    float* pos = out + (size_t)NB * NR * (NC * NP * NP) + (size_t)br * 2;
    pos[0] = (float)t;
    pos[1] = (float)l;
  }
}

extern "C" void kernel_launch(void* const* d_in, const int* in_sizes, int n_in,
                              void* d_out, int out_size, void* d_ws, size_t ws_size,
                              hipStream_t stream) {
  (void)in_sizes; (void)n_in; (void)d_ws; (void)ws_size; (void)out_size;
  const float* x  = (const float*)d_in[0];
  const int*   ct = (const int*)d_in[1];
  const int*   cl = (const int*)d_in[2];
  float*       o  = (float*)d_out;
  ImagePatcherOverlapRandom_65609920414052_kernel<<<NB * NR, 128, 0, stream>>>(x, ct, cl, o);
}